// DevConv_52896817217994
// MI455X (gfx1250) — compile-verified
//
#include <hip/hip_runtime.h>

typedef float v2f __attribute__((ext_vector_type(2)));
typedef float v8f __attribute__((ext_vector_type(8)));

// ---------------------------------------------------------------------------
// Kernel 1: zero the per-node max array (bits of +0.0f). Must run every call.
// ---------------------------------------------------------------------------
__global__ void init_max_kernel(unsigned int* __restrict__ maxbits, int n) {
    int i = blockIdx.x * blockDim.x + threadIdx.x;
    if (i < n) maxbits[i] = 0u;
}

// ---------------------------------------------------------------------------
// Kernel 2: mean(W_phi) over 128 elements using V_WMMA_F32_16X16X4_F32.
// A carries 64 W_phi values per WMMA (16x4 f32 = 2 VGPRs/lane), B = ones,
// so D[m][n] accumulates the row-sums; every column is identical.
// One wave (32 threads), EXEC all ones as WMMA requires.
// C/D layout: lane L, VGPR v holds (M = v + 8*(L>=16), N = L%16), so
// summing a lane's 8 components gives half a column; lane0 + lane16 = total.
// ---------------------------------------------------------------------------
__global__ void phi_mean_wmma_kernel(const float* __restrict__ W_phi,
                                     float* __restrict__ phi_mean) {
    int lane = threadIdx.x;  // 0..31, one full wave
    v2f a0, a1, b;
    a0.x = W_phi[lane];        a0.y = W_phi[lane + 32];
    a1.x = W_phi[lane + 64];   a1.y = W_phi[lane + 96];
    b.x = 1.0f; b.y = 1.0f;
    v8f c = {};
    c = __builtin_amdgcn_wmma_f32_16x16x4_f32(false, a0, false, b,
                                              (short)0, c, false, false);
    c = __builtin_amdgcn_wmma_f32_16x16x4_f32(false, a1, false, b,
                                              (short)0, c, false, false);
    float s = c[0] + c[1] + c[2] + c[3] + c[4] + c[5] + c[6] + c[7];
    float partner = __shfl(s, lane ^ 16, 32);  // lane0 gets lane16's half-column
    if (lane == 0) phi_mean[0] = (s + partner) * (1.0f / 128.0f);
}

// ---------------------------------------------------------------------------
// Kernel 3: per-edge |(nodes[r]-nodes[c]) . W_theta| scatter-max into maxbits.
// 4 edges per thread via int4 index loads (GLOBAL_LOAD_B128 on the two index
// streams). Distances >= 0, so uint atomicMax on float bits is an exact
// float max (lowers to non-returning GLOBAL_ATOMIC_MAX_U32, L2-resident).
// ---------------------------------------------------------------------------
__global__ void edge_max_kernel(const float* __restrict__ nodes,
                                const int* __restrict__ rows,
                                const int* __restrict__ cols,
                                const float* __restrict__ W_theta,
                                unsigned int* __restrict__ maxbits,
                                int n_edges) {
    const float w0 = W_theta[0], w1 = W_theta[1], w2 = W_theta[2];
    int base = (blockIdx.x * blockDim.x + threadIdx.x) * 4;
    if (base + 3 < n_edges) {
        int4 r4 = *reinterpret_cast<const int4*>(rows + base);
        int4 c4 = *reinterpret_cast<const int4*>(cols + base);
        int rr[4] = {r4.x, r4.y, r4.z, r4.w};
        int cc[4] = {c4.x, c4.y, c4.z, c4.w};
#pragma unroll
        for (int k = 0; k < 4; ++k) {
            const float* pa = nodes + 3 * rr[k];
            const float* pb = nodes + 3 * cc[k];
            float d = fabsf((pa[0] - pb[0]) * w0 +
                            (pa[1] - pb[1]) * w1 +
                            (pa[2] - pb[2]) * w2);
            atomicMax(maxbits + rr[k], __float_as_uint(d));
        }
    } else {
        for (int e = base; e < n_edges; ++e) {
            int r = rows[e], c = cols[e];
            const float* pa = nodes + 3 * r;
            const float* pb = nodes + 3 * c;
            float d = fabsf((pa[0] - pb[0]) * w0 +
                            (pa[1] - pb[1]) * w1 +
                            (pa[2] - pb[2]) * w2);
            atomicMax(maxbits + r, __float_as_uint(d));
        }
    }
}

// ---------------------------------------------------------------------------
// Kernel 4: out = 0.5 * (prev + maxdist * mean(W_phi))
// ---------------------------------------------------------------------------
__global__ void finalize_kernel(const float* __restrict__ prev,
                                const unsigned int* __restrict__ maxbits,
                                const float* __restrict__ phi_mean,
                                float* __restrict__ out, int n) {
    int i = blockIdx.x * blockDim.x + threadIdx.x;
    if (i < n) {
        float m = __uint_as_float(maxbits[i]);  // +0.0f for empty segments
        out[i] = 0.5f * (prev[i] + m * phi_mean[0]);
    }
}

extern "C" void kernel_launch(void* const* d_in, const int* in_sizes, int n_in,
                              void* d_out, int out_size, void* d_ws, size_t ws_size,
                              hipStream_t stream) {
    const float* prev    = (const float*)d_in[0];  // [N]
    const float* nodes   = (const float*)d_in[1];  // [N,3]
    const int*   rows    = (const int*)  d_in[2];  // [E]
    const int*   cols    = (const int*)  d_in[3];  // [E]
    const float* W_phi   = (const float*)d_in[4];  // [128]
    const float* W_theta = (const float*)d_in[5];  // [3,1]
    float*       out     = (float*)d_out;

    const int n_nodes = in_sizes[0];
    const int n_edges = in_sizes[2];

    // Workspace layout: [0, n_nodes) uint max-bits, then 1 float phi_mean.
    unsigned int* maxbits = (unsigned int*)d_ws;
    float* phi_mean = (float*)((unsigned int*)d_ws + n_nodes);

    const int B = 256;
    int node_blocks = (n_nodes + B - 1) / B;
    int edge_threads = (n_edges + 3) / 4;
    int edge_blocks = (edge_threads + B - 1) / B;

    init_max_kernel<<<node_blocks, B, 0, stream>>>(maxbits, n_nodes);
    phi_mean_wmma_kernel<<<1, 32, 0, stream>>>(W_phi, phi_mean);
    edge_max_kernel<<<edge_blocks, B, 0, stream>>>(nodes, rows, cols, W_theta,
                                                   maxbits, n_edges);
    finalize_kernel<<<node_blocks, B, 0, stream>>>(prev, maxbits, phi_mean,
                                                   out, n_nodes);
}